// Abstract_RC_31301721653455
// MI455X (gfx1250) — compile-verified
//
#include <hip/hip_runtime.h>
#include <hip/hip_bf16.h>

// ---------------------------------------------------------------------------
// Reservoir recurrence on MI455X (gfx1250).
//   feats_t = clip(W_res @ feats_{t-1} + x[:,t]);  out[:,t] = w_out @ feats_t
// Strategy: persistent kernel, one grid barrier per timestep; GEMV done with
// V_WMMA_F32_16X16X4_F32 (exact fp32) on pre-swizzled A-fragments; feats
// vector broadcast into all 16 B columns; out-GEMV pipelined 1 step behind.
// Weights (72 MB fp32) live in L2 (192 MB) across all 2048 steps.
// ---------------------------------------------------------------------------

#define FEAT     4096
#define ODIM     512
#define T_STEPS  2048

#define RES_TILES (FEAT / 16)            // 256
#define OUT_TILES (ODIM / 16)            // 32
#define N_TILES   (RES_TILES + OUT_TILES) // 288
#define KFRAG     (FEAT / 4)             // 1024 wmma k-fragments per tile
#define KSPLIT    4                      // waves per tile (K split 4 ways)
#define KF_PER_WAVE (KFRAG / KSPLIT)     // 256 wmma per wave per step
#define TPB       256                    // 8 waves
#define TILES_PER_BLOCK 2
#define NBLK      (N_TILES / TILES_PER_BLOCK) // 144

typedef __attribute__((ext_vector_type(2))) float v2f;
typedef __attribute__((ext_vector_type(4))) float v4f;
typedef __attribute__((ext_vector_type(8))) float v8f;

// ---------------------------------------------------------------------------
// Prepare: swizzle W_res / w_out rows into contiguous WMMA A-fragments.
// Fragment layout per (tile, j): 32 lanes x v2f (256B, coalesced).
//   lane l holds W[tile*16 + (l&15)][j*4 + 2*(l>>4) + {0,1}]
// Also zeros feats buffer 0 and the barrier counters.
// One thread per v2f:  tid = (tile*1024 + j)*32 + lane,  total 288*1024*32.
// ---------------------------------------------------------------------------
__global__ __launch_bounds__(TPB) void rc_prepare(
    const float* __restrict__ wres, const float* __restrict__ wout,
    v2f* __restrict__ wt, float* __restrict__ feats0,
    unsigned* __restrict__ syncv) {
  const unsigned tid = blockIdx.x * TPB + threadIdx.x;   // < 9,437,184 = 288<<15
  if (tid < FEAT) feats0[tid] = 0.0f;                    // feats (ping buffer) = 0
  if (tid == 0) { syncv[0] = 0u; syncv[1] = 0u; }        // barrier cnt / gen

  const unsigned lane = tid & 31u;
  const unsigned j    = (tid >> 5) & 1023u;
  const unsigned tg   = tid >> 15;                       // tile 0..287
  const unsigned m    = lane & 15u;
  const unsigned hi   = lane >> 4;
  const unsigned col  = j * 4u + 2u * hi;                // even -> 8B aligned
  const float* src;
  if (tg < RES_TILES) src = wres + (size_t)(tg * 16u + m) * FEAT + col;
  else                src = wout + (size_t)((tg - RES_TILES) * 16u + m) * FEAT + col;
  wt[tid] = *(const v2f*)src;                            // coalesced b64 store
}

// ---------------------------------------------------------------------------
// Device-scope grid barrier (all NBLK blocks resident).
// ---------------------------------------------------------------------------
__device__ __forceinline__ void grid_barrier(unsigned* cnt, unsigned* gen,
                                             unsigned nblk) {
  __syncthreads();
  if (threadIdx.x == 0) {
    __threadfence();
    unsigned g = __hip_atomic_load(gen, __ATOMIC_RELAXED, __HIP_MEMORY_SCOPE_AGENT);
    unsigned a = __hip_atomic_fetch_add(cnt, 1u, __ATOMIC_ACQ_REL,
                                        __HIP_MEMORY_SCOPE_AGENT);
    if (a == nblk - 1u) {
      __hip_atomic_store(cnt, 0u, __ATOMIC_RELAXED, __HIP_MEMORY_SCOPE_AGENT);
      __hip_atomic_fetch_add(gen, 1u, __ATOMIC_RELEASE, __HIP_MEMORY_SCOPE_AGENT);
    } else {
      while (__hip_atomic_load(gen, __ATOMIC_ACQUIRE,
                               __HIP_MEMORY_SCOPE_AGENT) == g) {
        __builtin_amdgcn_s_sleep(1);
      }
    }
  }
  __syncthreads();
}

// ---------------------------------------------------------------------------
// Persistent recurrence kernel. Block = 8 waves = 2 tiles x 4 K-chunks.
// Iteration i (0..T): res-tiles compute feats_{i} -> feats_{i+1} (skip i==T),
// out-tiles compute out[:, i-1] from feats_i (skip i==0). One barrier/step.
// ---------------------------------------------------------------------------
__global__ __launch_bounds__(TPB) void rc_persistent(
    const float* __restrict__ x,      // [FEAT][T]
    const v2f*   __restrict__ wt,     // swizzled fragments
    float*       __restrict__ feats,  // [2][FEAT] ping-pong
    float*       __restrict__ out,    // [ODIM][T]
    unsigned*    __restrict__ syncv) {
  __shared__ float ldsF[FEAT];                 // staged feats (16 KB)
  __shared__ float red[TILES_PER_BLOCK][KSPLIT][16];

  const int tid  = threadIdx.x;
  const int lane = tid & 31;
  const int wave = tid >> 5;
  const int tIB  = wave >> 2;                  // tile in block: 0..1
  const int kq   = wave & 3;                   // K chunk: 0..3
  const int tg   = blockIdx.x * TILES_PER_BLOCK + tIB;  // global tile
  const bool isRes = (tg < RES_TILES);
  const int hi   = lane >> 4;
  const int kbase = kq * (FEAT / KSPLIT);      // 0,1024,2048,3072

  // A-fragment stream for this wave: 256 coalesced b64 loads per step
  const v2f* __restrict__ ap =
      wt + ((size_t)tg * KFRAG + (size_t)kq * KF_PER_WAVE) * 32 + lane;
  // B (broadcast feats) from LDS: v2f at ldsF[kbase + 4j + 2*hi]
  const v2f* __restrict__ bp = (const v2f*)(ldsF + kbase) + hi;

  unsigned* cnt = syncv;
  unsigned* gen = syncv + 1;

  for (int i = 0; i <= T_STEPS; ++i) {
    const float* fcur  = feats + (i & 1) * FEAT;
    float*       fnext = feats + ((i + 1) & 1) * FEAT;

    // stage current feats vector into LDS (4 x b128 per thread)
    {
      const v4f* s = (const v4f*)fcur;
      v4f*       d = (v4f*)ldsF;
      for (int u = tid; u < FEAT / 4; u += TPB) d[u] = s[u];
    }
    __syncthreads();

    const bool active = isRes ? (i < T_STEPS) : (i > 0);
    if (active) {
      v8f c0 = {}, c1 = {};
#pragma unroll 4
      for (int j = 0; j < KF_PER_WAVE; j += 2) {
        v2f a0 = ap[(size_t)j * 32];
        v2f a1 = ap[(size_t)(j + 1) * 32];
        v2f b0 = bp[j * 2];
        v2f b1 = bp[j * 2 + 2];
        // D = A(16x4,f32) * B(4x16,f32 broadcast cols) + C   (exact fp32)
        c0 = __builtin_amdgcn_wmma_f32_16x16x4_f32(false, a0, false, b0,
                                                   (short)0, c0, false, false);
        c1 = __builtin_amdgcn_wmma_f32_16x16x4_f32(false, a1, false, b1,
                                                   (short)0, c1, false, false);
      }
      v8f c = c0 + c1;
      // all 16 columns identical: lane 0 holds rows m0+0..7 (vgpr r),
      // lane 16 holds rows m0+8..15.
      if (lane == 0) {
#pragma unroll
        for (int r = 0; r < 8; ++r) red[tIB][kq][r] = c[r];
      } else if (lane == 16) {
#pragma unroll
        for (int r = 0; r < 8; ++r) red[tIB][kq][8 + r] = c[r];
      }
    }
    __syncthreads();

    // finalize: reduce 4 K-chunk partials, apply x + clip (res) or store out
    if (tid < 32) {
      const int t2 = tid >> 4;
      const int r  = tid & 15;
      const int tg2 = blockIdx.x * TILES_PER_BLOCK + t2;
      if (tg2 < RES_TILES) {
        if (i < T_STEPS) {
          float s = red[t2][0][r] + red[t2][1][r] + red[t2][2][r] + red[t2][3][r];
          const int fr = tg2 * 16 + r;
          s += x[(size_t)fr * T_STEPS + i];
          s = fminf(1.0f, fmaxf(-1.0f, s));
          fnext[fr] = s;
        }
      } else {
        if (i > 0) {
          float s = red[t2][0][r] + red[t2][1][r] + red[t2][2][r] + red[t2][3][r];
          const int orow = (tg2 - RES_TILES) * 16 + r;
          out[(size_t)orow * T_STEPS + (i - 1)] = s;
        }
      }
    }
    grid_barrier(cnt, gen, gridDim.x);
  }
}

// ---------------------------------------------------------------------------
// ws layout (floats): [0, 18874368)            swizzled fragments (72 MB)
//                     [+0, +8192)              feats ping-pong (32 KB)
//                     [+8192, +8194)           barrier cnt/gen (as uint)
// ---------------------------------------------------------------------------
extern "C" void kernel_launch(void* const* d_in, const int* in_sizes, int n_in,
                              void* d_out, int out_size, void* d_ws, size_t ws_size,
                              hipStream_t stream) {
  const float* x    = (const float*)d_in[0];  // [4096][2048]
  const float* wres = (const float*)d_in[1];  // [4096][4096]
  const float* wout = (const float*)d_in[2];  // [512][4096]
  float* out = (float*)d_out;                 // [512][2048]

  float* wt    = (float*)d_ws;                            // 288*1024*64 floats
  float* feats = wt + (size_t)N_TILES * KFRAG * 64;       // 2*4096 floats
  unsigned* syncv = (unsigned*)(feats + 2 * FEAT);

  const int prep_threads = N_TILES * KFRAG * 32;          // one per v2f
  rc_prepare<<<prep_threads / TPB, TPB, 0, stream>>>(wres, wout, (v2f*)wt,
                                                     feats, syncv);
  rc_persistent<<<NBLK, TPB, 0, stream>>>(x, (const v2f*)wt, feats, out, syncv);
}